// GCNRegressor_47725676593414
// MI455X (gfx1250) — compile-verified
//
#include <hip/hip_runtime.h>

typedef __attribute__((ext_vector_type(2))) float v2f;
typedef __attribute__((ext_vector_type(8))) float v8f;

#define HID 64

// ---------------------------------------------------------------------------
// deg[i] = 1.0 (self-loop weight), then atomic scatter of edge weights,
// then deg -> rsqrt(deg) in place.
// ---------------------------------------------------------------------------
__global__ void k_fill1(float* __restrict__ p, int n) {
  int i = blockIdx.x * blockDim.x + threadIdx.x;
  if (i < n) p[i] = 1.0f;
}

__global__ void k_deg_scatter(const int* __restrict__ dst,
                              const float* __restrict__ ew,
                              float* __restrict__ deg, int e_cnt) {
  int e = blockIdx.x * blockDim.x + threadIdx.x;
  if (e < e_cnt) atomicAdd(&deg[dst[e]], ew[e]);
}

__global__ void k_rsqrt(float* __restrict__ p, int n) {
  int i = blockIdx.x * blockDim.x + threadIdx.x;
  if (i < n) {
    float d = p[i];
    p[i] = d > 0.0f ? rsqrtf(d) : 0.0f;
  }
}

// ---------------------------------------------------------------------------
// C[nrows x 64] = op(A)[nrows x KDIM] @ B[KDIM x 64], op = relu if RELU_IN.
// B is staged once per workgroup into LDS via GLOBAL_LOAD_ASYNC_TO_LDS_B128
// (ASYNCcnt path), then WMMA B-fragments are fed from ds_load.
// One wave per 16-row tile; 4 accumulators cover all 64 output columns.
// Uses V_WMMA_F32_16X16X4_F32 (exact f32): A 16x4, B 4x16, C 16x16.
//   A frag: lane=(half,m): a[v] = A[row0+m][k + 2*half + v]   (float2 load)
//   B frag: b[v] = B[k + 2*half + v][col0 + m]                (LDS)
//   C frag: c[v] -> row (row0 + v + 8*half), col (col0 + m)
// ---------------------------------------------------------------------------
template <int KDIM, bool RELU_IN>
__global__ void k_gemm_wmma(const float* __restrict__ A,
                            const float* __restrict__ B,
                            float* __restrict__ C, int nrows) {
  __shared__ float sB[KDIM * HID];

  // --- cooperative async stage of B into LDS (all threads participate) ---
  {
    unsigned lds_base = (unsigned)(size_t)(&sB[0]);
    const int nchunk = (KDIM * HID) / 4;  // 16-byte chunks
    for (int c = threadIdx.x; c < nchunk; c += blockDim.x) {
      unsigned lds_addr = lds_base + (unsigned)c * 16u;
      const float* gaddr = B + (size_t)c * 4;
      asm volatile("global_load_async_to_lds_b128 %0, %1, off"
                   :
                   : "v"(lds_addr), "v"(gaddr)
                   : "memory");
    }
    asm volatile("s_wait_asynccnt 0x0" ::: "memory");
  }
  __syncthreads();

  const int gtid = blockIdx.x * blockDim.x + threadIdx.x;
  const int wave = gtid >> 5;
  const int lane = threadIdx.x & 31;
  const int ntile = (nrows + 15) >> 4;
  if (wave >= ntile) return;  // wave-uniform: EXEC stays all-ones for WMMA

  const int row0 = wave << 4;
  const int m    = lane & 15;
  const int half = lane >> 4;

  int arow_i = row0 + m;
  if (arow_i >= nrows) arow_i = nrows - 1;  // clamp (EXEC must stay all-ones)
  const float* __restrict__ arow = A + (size_t)arow_i * KDIM;

  v8f c0 = {}, c1 = {}, c2 = {}, c3 = {};

  for (int k = 0; k < KDIM; k += 4) {
    const int ka = k + 2 * half;
    float a0 = arow[ka];
    float a1 = arow[ka + 1];
    if (RELU_IN) {
      a0 = fmaxf(a0, 0.0f);
      a1 = fmaxf(a1, 0.0f);
    }
    v2f a;
    a.x = a0;
    a.y = a1;

    const float* brow = &sB[ka * HID + m];
    v2f b0, b1, b2, b3;
    b0.x = brow[0];       b0.y = brow[HID + 0];
    b1.x = brow[16];      b1.y = brow[HID + 16];
    b2.x = brow[32];      b2.y = brow[HID + 32];
    b3.x = brow[48];      b3.y = brow[HID + 48];

    c0 = __builtin_amdgcn_wmma_f32_16x16x4_f32(false, a, false, b0, (short)0, c0, false, false);
    c1 = __builtin_amdgcn_wmma_f32_16x16x4_f32(false, a, false, b1, (short)0, c1, false, false);
    c2 = __builtin_amdgcn_wmma_f32_16x16x4_f32(false, a, false, b2, (short)0, c2, false, false);
    c3 = __builtin_amdgcn_wmma_f32_16x16x4_f32(false, a, false, b3, (short)0, c3, false, false);
  }

#pragma unroll
  for (int v = 0; v < 8; ++v) {
    int rr = row0 + v + 8 * half;
    if (rr < nrows) {
      float* crow = C + (size_t)rr * HID + m;
      crow[0]  = c0[v];
      crow[16] = c1[v];
      crow[32] = c2[v];
      crow[48] = c3[v];
    }
  }
}

// ---------------------------------------------------------------------------
// agg[i][j] = bias[j] + dinv[i]^2 * h[i][j]   (bias + self-loop message)
// ---------------------------------------------------------------------------
__global__ void k_agg_init(float* __restrict__ agg,
                           const float* __restrict__ bias,
                           const float* __restrict__ dinv,
                           const float* __restrict__ h, int n) {
  int t = blockIdx.x * blockDim.x + threadIdx.x;
  if (t >= n * HID) return;
  int i = t >> 6;
  int j = t & 63;
  float di = dinv[i];
  agg[t] = bias[j] + di * di * h[t];
}

// ---------------------------------------------------------------------------
// One wave per edge: 32 lanes x float2 = full 64-wide feature row.
// agg[dst] += (dinv[src]*ew*dinv[dst]) * h[src]  via f32 atomics.
// ---------------------------------------------------------------------------
__global__ void k_edge_scatter(const int* __restrict__ src,
                               const int* __restrict__ dst,
                               const float* __restrict__ ew,
                               const float* __restrict__ dinv,
                               const float* __restrict__ h,
                               float* __restrict__ agg, int e_cnt) {
  int t = blockIdx.x * blockDim.x + threadIdx.x;
  int e = t >> 5;
  int lane = t & 31;
  if (e >= e_cnt) return;
  int s = src[e];
  int d = dst[e];
  float w = dinv[s] * ew[e] * dinv[d];
  const float2* __restrict__ hv = (const float2*)(h + (size_t)s * HID);
  float2 v = hv[lane];
  float* ap = agg + (size_t)d * HID + 2 * lane;
  atomicAdd(ap,     w * v.x);
  atomicAdd(ap + 1, w * v.y);
}

// ---------------------------------------------------------------------------
// out[i] = relu(agg[i][:]) . Wl + bl
// ---------------------------------------------------------------------------
__global__ void k_final(const float* __restrict__ agg,
                        const float* __restrict__ Wl,
                        const float* __restrict__ bl,
                        float* __restrict__ out, int n) {
  int i = blockIdx.x * blockDim.x + threadIdx.x;
  if (i >= n) return;
  const float* row = agg + (size_t)i * HID;
  float acc = bl[0];
#pragma unroll
  for (int j = 0; j < HID; ++j) {
    acc = fmaf(fmaxf(row[j], 0.0f), Wl[j], acc);
  }
  out[i] = acc;
}

// ---------------------------------------------------------------------------
extern "C" void kernel_launch(void* const* d_in, const int* in_sizes, int n_in,
                              void* d_out, int out_size, void* d_ws, size_t ws_size,
                              hipStream_t stream) {
  const float* x  = (const float*)d_in[0];
  const int*   ei = (const int*)d_in[1];
  const float* ew = (const float*)d_in[2];
  const float* W1 = (const float*)d_in[3];
  const float* b1 = (const float*)d_in[4];
  const float* W2 = (const float*)d_in[5];
  const float* b2 = (const float*)d_in[6];
  const float* Wl = (const float*)d_in[7];
  const float* bl = (const float*)d_in[8];

  const int N = in_sizes[0] / 128;  // IN_DIM = 128
  const int E = in_sizes[1] / 2;
  const int* src = ei;
  const int* dst = ei + E;

  float* ws   = (float*)d_ws;
  float* dinv = ws;                          // N floats (deg -> dinv in place)
  float* bufA = ws + N;                      // N*64: X@W1, later relu(agg1)@W2
  float* bufB = bufA + (size_t)N * HID;      // N*64: agg1, later agg2
  float* out  = (float*)d_out;

  const int BLK = 256;
  const int gN    = (N + BLK - 1) / BLK;
  const int gE    = (E + BLK - 1) / BLK;
  const int gNH   = (N * HID + BLK - 1) / BLK;
  const int gEdgW = (int)(((long long)E * 32 + BLK - 1) / BLK);
  const int ntile = (N + 15) / 16;
  const int gGemm = (int)(((long long)ntile * 32 + BLK - 1) / BLK);

  // normalization: deg = 1 + sum_in(ew); dinv = rsqrt(deg)
  k_fill1<<<gN, BLK, 0, stream>>>(dinv, N);
  k_deg_scatter<<<gE, BLK, 0, stream>>>(dst, ew, dinv, E);
  k_rsqrt<<<gN, BLK, 0, stream>>>(dinv, N);

  // layer 1: bufA = X@W1 ; bufB = b1 + dinv^2*bufA + scatter(edges)
  k_gemm_wmma<128, false><<<gGemm, BLK, 0, stream>>>(x, W1, bufA, N);
  k_agg_init<<<gNH, BLK, 0, stream>>>(bufB, b1, dinv, bufA, N);
  k_edge_scatter<<<gEdgW, BLK, 0, stream>>>(src, dst, ew, dinv, bufA, bufB, E);

  // layer 2: bufA = relu(bufB)@W2 ; bufB = b2 + dinv^2*bufA + scatter(edges)
  k_gemm_wmma<64, true><<<gGemm, BLK, 0, stream>>>(bufB, W2, bufA, N);
  k_agg_init<<<gNH, BLK, 0, stream>>>(bufB, b2, dinv, bufA, N);
  k_edge_scatter<<<gEdgW, BLK, 0, stream>>>(src, dst, ew, dinv, bufA, bufB, E);

  // head: out = relu(bufB) @ Wl + bl
  k_final<<<gN, BLK, 0, stream>>>(bufB, Wl, bl, out, N);
}